// MixtureGaussianSequenceLabeling_31619549233747
// MI455X (gfx1250) — compile-verified
//
#include <hip/hip_runtime.h>
#include <math.h>

// ---------------------------------------------------------------------------
// MixtureGaussianSequenceLabeling on MI455X (gfx1250, wave32)
//   kernel 1 (scan):  BATCH x 2 blocks, K-best Gaussian message passing.
//                     32x32 / 64x64 f32 matmuls via V_WMMA_F32_16X16X4_F32.
//   kernel 2 (merge): BATCH x MAXLEN blocks, fwd(+)input(+)bwd merge, top-2,
//                     label emission scores + logsumexp.
//   History (8.7 MB) staged in d_ws -> stays L2-resident (192 MB L2).
// ---------------------------------------------------------------------------

#define G_DIM     32
#define G_D2      64
#define G_IC      2
#define G_TC      2
#define G_K       2
#define G_BATCH   16
#define G_MAXLEN  32
#define G_NLABELS 20
#define LOG2PI    1.8378770664093453f

#define BLK 128   // 4 wave32 waves == the 4 SIMD32s of one WGP

typedef float v2f __attribute__((ext_vector_type(2)));
typedef float v8f __attribute__((ext_vector_type(8)));

// hypothesis slot layout (scan kernel), floats
#define SLOT     2116
#define LAM_OFF  0
#define VAR_OFF  1024
#define ETA_OFF  2048
#define MU_OFF   2080
#define SC_OFF   2112
#define LD_OFF   2113

#define HIST_ESZ 1057   // score + mu(32) + var(32x32)

static constexpr int SCAN_LDS_FLOATS  = 22400;  // carved: 22336
static constexpr int MERGE_LDS_FLOATS = 11072;  // carved: 11020

__device__ __forceinline__ size_t hist_idx(int b, int dir, int t, int k) {
  return ((size_t)(((b * 2 + dir) * G_MAXLEN + t) * G_K + k)) * HIST_ESZ;
}

// ---------------------------------------------------------------------------
// block-cooperative primitives (blockDim.x == BLK assumed)
// ---------------------------------------------------------------------------
__device__ __forceinline__ void bcopy(float* dst, const float* src, int n) {
  for (int i = threadIdx.x; i < n; i += BLK) dst[i] = src[i];
}

__device__ __forceinline__ float block_sum(float p, float* red) {
  red[threadIdx.x] = p;
  __syncthreads();
  for (int s = BLK >> 1; s > 0; s >>= 1) {
    if ((int)threadIdx.x < s) red[threadIdx.x] += red[threadIdx.x + s];
    __syncthreads();
  }
  float r = red[0];
  __syncthreads();
  return r;
}

__device__ __forceinline__ float block_dot(const float* a, const float* b,
                                           int n, float* red) {
  float p = 0.f;
  for (int i = threadIdx.x; i < n; i += BLK) p += a[i] * b[i];
  return block_sum(p, red);
}

__device__ __forceinline__ void matvec(const float* M, int ld, const float* x,
                                       float* y, int n) {
  for (int i = threadIdx.x; i < n; i += BLK) {
    float s = 0.f;
    for (int j = 0; j < n; ++j) s += M[i * ld + j] * x[j];
    y[i] = s;
  }
  __syncthreads();
}

// ---------------------------------------------------------------------------
// C = alpha * op(A)op(B) [+ C0], n x n with n in {32, 64}.
// Uses V_WMMA_F32_16X16X4_F32: per wave, one 16x16 f32 tile, K chained in
// steps of 4. Fragment layouts per CDNA5 ISA 7.12.2:
//   A 16x4  : lane 0..15 M=lane, VGPR j holds K = j + 2*(lane>>4)
//   B 4x16  : lane 0..15 N=lane, VGPR j holds K = j + 2*(lane>>4)
//   C 16x16 : VGPR v holds row M = v + 8*(lane>>4), col N = lane&15
// All 32 lanes of every wave are active at every call site (EXEC == all 1s).
// ---------------------------------------------------------------------------
__device__ __forceinline__ void mm_wmma(int n, const float* A, int lda, bool tA,
                                        const float* B, int ldb, bool tB,
                                        float* C, int ldc, float alpha,
                                        const float* C0, int ldc0) {
  const int lane = threadIdx.x & 31;
  const int wave = threadIdx.x >> 5;
  const int half = lane >> 4;
  const int l15  = lane & 15;
  const int nt   = n >> 4;
  for (int tile = wave; tile < nt * nt; tile += 4) {
    const int ti = tile / nt;
    const int tj = tile % nt;
    v8f acc = {0.f, 0.f, 0.f, 0.f, 0.f, 0.f, 0.f, 0.f};
    for (int k0 = 0; k0 < n; k0 += 4) {
      v2f a = {0.f, 0.f};
      v2f bb = {0.f, 0.f};
#pragma unroll
      for (int j = 0; j < 2; ++j) {
        const int k  = k0 + j + 2 * half;
        const int m  = ti * 16 + l15;
        const int nn = tj * 16 + l15;
        a[j]  = tA ? A[k * lda + m] : A[m * lda + k];
        bb[j] = tB ? B[nn * ldb + k] : B[k * ldb + nn];
      }
      acc = __builtin_amdgcn_wmma_f32_16x16x4_f32(false, a, false, bb,
                                                  (short)0, acc, false, false);
    }
#pragma unroll
    for (int v = 0; v < 8; ++v) {
      const int r  = ti * 16 + v + 8 * half;
      const int cc = tj * 16 + l15;
      float out = alpha * acc[v];
      if (C0) out += C0[r * ldc0 + cc];
      C[r * ldc + cc] = out;
    }
  }
}

// ---------------------------------------------------------------------------
// In-place Gauss-Jordan: destroys M (n x n, LDS), writes Inv = M^-1,
// *ldac = log|det M|. Pivot column cached in fcol to avoid races.
// ---------------------------------------------------------------------------
__device__ void gj_inv(float* M, float* Inv, int n, float* fcol, float* ldac) {
  const int tid = threadIdx.x;
  for (int i = tid; i < n * n; i += BLK)
    Inv[i] = ((i / n) == (i % n)) ? 1.f : 0.f;
  if (tid == 0) *ldac = 0.f;
  __syncthreads();
  for (int k = 0; k < n; ++k) {
    const float piv = M[k * n + k];
    __syncthreads();
    const float s = 1.f / piv;
    for (int j = tid; j < n; j += BLK) {
      M[k * n + j] *= s;
      Inv[k * n + j] *= s;
    }
    for (int i = tid; i < n; i += BLK) fcol[i] = (i == k) ? 0.f : M[i * n + k];
    if (tid == 0) *ldac += logf(fabsf(piv) + 1e-30f);
    __syncthreads();
    for (int idx = tid; idx < n * n; idx += BLK) {
      const int i = idx / n;
      if (i == k) continue;
      const int j = idx - i * n;
      const float f = fcol[i];
      M[idx]   -= f * M[k * n + j];
      Inv[idx] -= f * Inv[k * n + j];
    }
    __syncthreads();
  }
}

// ---------------------------------------------------------------------------
// Scan kernel context (pointers carved from dynamic LDS)
// ---------------------------------------------------------------------------
struct ScanCtx {
  float *lamT, *etaT, *zetaT;
  float *w0, *w1, *w2, *w3;     // contiguous 4x1024 scratch (also GJ64 M)
  float *LF, *eta_f, *e1, *ae1, *etaM, *muM, *dinv, *etaIn;
  float *fcol, *red, *spool;    // spool: 0 tmp, 1 ldA, 2 ldM, 3 base
  int   *ctrl;
};

__device__ void write_hist(const float* cur, float* hist, int b, int dir,
                           int t) {
  for (int k = 0; k < G_K; ++k) {
    const size_t base = hist_idx(b, dir, t, k);
    const float* s = cur + k * SLOT;
    if (threadIdx.x == 0) hist[base] = s[SC_OFF];
    for (int i = threadIdx.x; i < 32; i += BLK) hist[base + 1 + i] = s[MU_OFF + i];
    for (int i = threadIdx.x; i < 1024; i += BLK)
      hist[base + 33 + i] = s[VAR_OFF + i];
  }
  __syncthreads();
}

// One transition candidate: marginalize 64-dim joint (lamT[c] + folded
// 32-dim message) down to a 32-dim message via Schur complement; score and
// insert into the running top-2 (nxt).
__device__ void do_transition(const ScanCtx& S, int c, float* nxt) {
  const int tid = threadIdx.x;
  const float* Lt = S.lamT + c * 4096;

  // A = lamT[c][0:32,0:32] + LF ; e1 = etaT[c][:32] + eta_f
  for (int idx = tid; idx < 1024; idx += BLK) {
    const int r = idx >> 5, q = idx & 31;
    S.w0[idx] = Lt[r * 64 + q] + S.LF[idx];
  }
  for (int i = tid; i < 32; i += BLK) S.e1[i] = S.etaT[c * 64 + i] + S.eta_f[i];
  __syncthreads();

  gj_inv(S.w0, S.w1, 32, S.fcol, &S.spool[1]);         // Ainv=w1, ldA
  matvec(S.w1, 32, S.e1, S.ae1, 32);                   // Ainv*e1
  const float qA = block_dot(S.e1, S.ae1, 32, S.red);
  const float zA = 0.5f * (qA + 32.f * LOG2PI - S.spool[1]);

  // AinvB = Ainv * B                   (WMMA, B = lamT[:32,32:], ld 64)
  mm_wmma(32, S.w1, 32, false, Lt + 32, 64, false, S.w2, 32, 1.f, nullptr, 0);
  __syncthreads();
  // lam_m = C - B^T * AinvB            (WMMA, C = lamT[32:,32:], ld 64)
  mm_wmma(32, Lt + 32, 64, true, S.w2, 32, false, S.w3, 32, -1.f,
          Lt + 32 * 64 + 32, 64);
  __syncthreads();

  // eta_m = e2 - B^T * (Ainv e1)
  for (int i = tid; i < 32; i += BLK) {
    float sacc = 0.f;
    for (int r = 0; r < 32; ++r) sacc += Lt[r * 64 + 32 + i] * S.ae1[r];
    S.etaM[i] = S.etaT[c * 64 + 32 + i] - sacc;
  }
  __syncthreads();

  // var_m = inv(lam_m), mu_m = var_m * eta_m
  bcopy(S.w0, S.w3, 1024);
  __syncthreads();
  gj_inv(S.w0, S.w1, 32, S.fcol, &S.spool[2]);         // var_m=w1, ldM
  matvec(S.w1, 32, S.etaM, S.muM, 32);
  const float qM = block_dot(S.etaM, S.muM, 32, S.red);
  const float zM = 0.5f * (qM + 32.f * LOG2PI - S.spool[2]);

  const float stot = S.spool[3] + zM + zA - S.zetaT[c];

  // running top-2 insert
  if (tid == 0) {
    int slot = -1;
    if (stot > nxt[SC_OFF]) slot = 0;
    else if (stot > nxt[SLOT + SC_OFF]) slot = 1;
    S.ctrl[0] = slot;
  }
  __syncthreads();
  const int slot = S.ctrl[0];
  if (slot < 0) return;
  if (slot == 0) {
    bcopy(nxt + SLOT, nxt, SLOT);   // shift best -> second
    __syncthreads();
  }
  float* d = nxt + slot * SLOT;
  bcopy(d + LAM_OFF, S.w3, 1024);
  bcopy(d + VAR_OFF, S.w1, 1024);
  bcopy(d + ETA_OFF, S.etaM, 32);
  bcopy(d + MU_OFF, S.muM, 32);
  if (tid == 0) { d[SC_OFF] = stot; d[LD_OFF] = S.spool[2]; }
  __syncthreads();
}

// ---------------------------------------------------------------------------
// Kernel 1: forward/backward K-best scan. grid (BATCH, 2), block 128.
// ---------------------------------------------------------------------------
__global__ __launch_bounds__(BLK) void mgsl_scan_kernel(
    const int* __restrict__ sent, const float* __restrict__ imu_emb,
    const float* __restrict__ icho_emb, const float* __restrict__ tmu,
    const float* __restrict__ tcho, float* __restrict__ hist) {
  extern __shared__ float lds[];
  ScanCtx S;
  float* p = lds;
  S.lamT  = p; p += G_TC * G_D2 * G_D2;
  S.etaT  = p; p += G_TC * G_D2;
  S.zetaT = p; p += 4;
  S.w0 = p; p += 1024;
  S.w1 = p; p += 1024;
  S.w2 = p; p += 1024;
  S.w3 = p; p += 1024;
  S.LF    = p; p += 1024;
  S.eta_f = p; p += 32;
  S.e1    = p; p += 32;
  S.ae1   = p; p += 32;
  S.etaM  = p; p += 32;
  S.muM   = p; p += 32;
  S.dinv  = p; p += 32;
  S.etaIn = p; p += 32;
  float* hypA = p; p += G_K * SLOT;
  float* hypB = p; p += G_K * SLOT;
  S.fcol  = p; p += 64;
  S.red   = p; p += BLK;
  S.spool = p; p += 8;
  S.ctrl  = (int*)p;

  const int b = blockIdx.x, dir = blockIdx.y, tid = threadIdx.x;

  // Phase 1: trans_var[c] = cho^T cho (WMMA 64x64), lamT = inv, etaT, zetaT.
  for (int c = 0; c < G_TC; ++c) {
    mm_wmma(64, tcho + c * 4096, 64, true, tcho + c * 4096, 64, false, S.w0,
            64, 1.f, nullptr, 0);
    __syncthreads();
    gj_inv(S.w0, S.lamT + c * 4096, 64, S.fcol, &S.spool[0]);
    matvec(S.lamT + c * 4096, 64, tmu + c * 64, S.etaT + c * 64, 64);
    const float quad = block_dot(S.etaT + c * 64, tmu + c * 64, 64, S.red);
    if (tid == 0) S.zetaT[c] = 0.5f * (quad + 64.f * LOG2PI + S.spool[0]);
    __syncthreads();
  }

  // Init hypotheses: fold Gaussian is N(0, I)  ->  base = -zeta(0, I).
  float* cur = hypA;
  float* nxt = hypB;
  if (tid < G_K) nxt[tid * SLOT + SC_OFF] = -3.0e38f;
  for (int i = tid; i < 1024; i += BLK)
    S.LF[i] = ((i >> 5) == (i & 31)) ? 1.f : 0.f;
  for (int i = tid; i < 32; i += BLK) S.eta_f[i] = 0.f;
  if (tid == 0) S.spool[3] = -16.f * LOG2PI;
  __syncthreads();
  for (int c = 0; c < G_TC; ++c) do_transition(S, c, nxt);
  { float* t = cur; cur = nxt; nxt = t; }
  write_hist(cur, hist, b, dir, 0);

  // Phase 2: scan tokens (history states 1..MAXLEN-1).
  for (int j = 0; j < G_MAXLEN - 1; ++j) {
    const int tok = sent[b * G_MAXLEN + (dir ? (G_MAXLEN - 1 - j) : j)];
    if (j + 1 < G_MAXLEN - 1) {
      const int tok2 =
          sent[b * G_MAXLEN + (dir ? (G_MAXLEN - 2 - j) : (j + 1))];
      __builtin_prefetch(&imu_emb[(size_t)tok2 * (G_IC * G_DIM)], 0, 1);
      __builtin_prefetch(&icho_emb[(size_t)tok2 * (G_IC * G_DIM)], 0, 1);
    }
    if (tid < G_K) nxt[tid * SLOT + SC_OFF] = -3.0e38f;
    __syncthreads();
    for (int k = 0; k < G_K; ++k) {
      const float* h = cur + k * SLOT;
      for (int i = 0; i < G_IC; ++i) {
        const float* gm = imu_emb + (size_t)tok * (G_IC * G_DIM) + i * G_DIM;
        const float* gc = icho_emb + (size_t)tok * (G_IC * G_DIM) + i * G_DIM;
        for (int q = tid; q < 32; q += BLK) {
          const float ch = gc[q];
          const float dv = ch * ch;
          S.dinv[q]  = 1.f / dv;
          S.etaIn[q] = gm[q] / dv;
        }
        __syncthreads();
        const float qin = block_dot(S.etaIn, gm, 32, S.red);
        float plg = 0.f;
        for (int q = tid; q < 32; q += BLK) plg += -logf(S.dinv[q]);
        const float sld = block_sum(plg, S.red);
        const float zin = 0.5f * (qin + 32.f * LOG2PI + sld);
        // folded message: lam_f = lam_prev + diag(1/d), eta_f = eta_prev+mu/d
        for (int idx = tid; idx < 1024; idx += BLK) {
          const int r = idx >> 5, q2 = idx & 31;
          S.LF[idx] = h[LAM_OFF + idx] + ((r == q2) ? S.dinv[r] : 0.f);
        }
        for (int q = tid; q < 32; q += BLK)
          S.eta_f[q] = h[ETA_OFF + q] + S.etaIn[q];
        __syncthreads();
        // zeta(fold) cancels against the integral's -zeta(fold):
        // base = prev_score - zeta(prev) - zeta(input)
        const float qp = block_dot(h + ETA_OFF, h + MU_OFF, 32, S.red);
        const float zp = 0.5f * (qp + 32.f * LOG2PI - h[LD_OFF]);
        if (tid == 0) S.spool[3] = h[SC_OFF] - zp - zin;
        __syncthreads();
        for (int c = 0; c < G_TC; ++c) do_transition(S, c, nxt);
      }
    }
    { float* t = cur; cur = nxt; nxt = t; }
    write_hist(cur, hist, b, dir, j + 1);
  }
}

// ---------------------------------------------------------------------------
// Kernel 2: merge fwd x input x bwd, top-2, emission + logsumexp.
// grid (BATCH, MAXLEN), block 128.
// ---------------------------------------------------------------------------
__global__ __launch_bounds__(BLK) void mgsl_merge_kernel(
    const int* __restrict__ sent, const float* __restrict__ imu_emb,
    const float* __restrict__ icho_emb, const float* __restrict__ omu,
    const float* __restrict__ ocho, const float* __restrict__ hist,
    float* __restrict__ out) {
  extern __shared__ float lds[];
  float* p = lds;
  float* pl    = p; p += 4 * 1024;   // fold precisions
  float* pe    = p; p += 4 * 32;     // fold etas
  float* pps   = p; p += 4;          // fold partial scores
  float* bl    = p; p += 2 * 1024;   // backward precisions
  float* be    = p; p += 2 * 32;     // backward etas
  float* bz    = p; p += 2;          // backward zetas
  float* bnbs  = p; p += 2;          // backward scores
  float* w0    = p; p += 1024;
  float* w1    = p; p += 1024;
  float* cand  = p; p += 2 * 1060;   // var(1024), mu(32), score
  float* etaE  = p; p += 32;
  float* muE   = p; p += 32;
  float* dinv  = p; p += 32;
  float* etaIn = p; p += 32;
  float* eta0  = p; p += 32;
  float* diff  = p; p += 32;
  float* sol   = p; p += 32;
  float* sc80  = p; p += 80;
  float* fcol  = p; p += 64;
  float* red   = p; p += BLK;
  float* spool = p; p += 8;
  int*   ctrl  = (int*)p;

  const int b = blockIdx.x, t = blockIdx.y, tid = threadIdx.x;
  const int tok = sent[b * G_MAXLEN + t];

  // backward suffix messages (mirror index)
  for (int k2 = 0; k2 < G_K; ++k2) {
    const size_t hb = hist_idx(b, 1, G_MAXLEN - 1 - t, k2);
    bcopy(w0, hist + hb + 33, 1024);
    __syncthreads();
    gj_inv(w0, bl + k2 * 1024, 32, fcol, &spool[0]);
    matvec(bl + k2 * 1024, 32, hist + hb + 1, be + k2 * 32, 32);
    const float q = block_dot(be + k2 * 32, hist + hb + 1, 32, red);
    if (tid == 0) {
      bz[k2]   = 0.5f * (q + 32.f * LOG2PI - spool[0]);
      bnbs[k2] = hist[hb];
    }
    __syncthreads();
  }

  // forward prefix + input-component folds -> 4 partial posteriors
  int pc = 0;
  for (int k1 = 0; k1 < G_K; ++k1) {
    const size_t hf = hist_idx(b, 0, t, k1);
    bcopy(w0, hist + hf + 33, 1024);
    __syncthreads();
    gj_inv(w0, w1, 32, fcol, &spool[0]);   // lam0 = w1
    matvec(w1, 32, hist + hf + 1, eta0, 32);
    const float q0 = block_dot(eta0, hist + hf + 1, 32, red);
    const float z0 = 0.5f * (q0 + 32.f * LOG2PI - spool[0]);
    const float fs = hist[hf];
    for (int i = 0; i < G_IC; ++i) {
      const float* gm = imu_emb + (size_t)tok * (G_IC * G_DIM) + i * G_DIM;
      const float* gc = icho_emb + (size_t)tok * (G_IC * G_DIM) + i * G_DIM;
      for (int q = tid; q < 32; q += BLK) {
        const float ch = gc[q];
        const float dv = ch * ch;
        dinv[q]  = 1.f / dv;
        etaIn[q] = gm[q] / dv;
      }
      __syncthreads();
      const float qin = block_dot(etaIn, gm, 32, red);
      float plg = 0.f;
      for (int q = tid; q < 32; q += BLK) plg += -logf(dinv[q]);
      const float sld = block_sum(plg, red);
      const float zin = 0.5f * (qin + 32.f * LOG2PI + sld);
      for (int idx = tid; idx < 1024; idx += BLK) {
        const int r = idx >> 5, q2 = idx & 31;
        pl[pc * 1024 + idx] = w1[idx] + ((r == q2) ? dinv[r] : 0.f);
      }
      for (int q = tid; q < 32; q += BLK) pe[pc * 32 + q] = eta0[q] + etaIn[q];
      if (tid == 0) pps[pc] = fs - z0 - zin;   // zeta(fold) cancels later
      __syncthreads();
      ++pc;
    }
  }

  // 4 x 2 merge candidates, keep top-2
  if (tid < G_K) cand[tid * 1060 + 1056] = -3.0e38f;
  __syncthreads();
  for (int pq = 0; pq < 4; ++pq) {
    for (int k2 = 0; k2 < G_K; ++k2) {
      for (int idx = tid; idx < 1024; idx += BLK)
        w0[idx] = pl[pq * 1024 + idx] + bl[k2 * 1024 + idx];
      for (int q = tid; q < 32; q += BLK)
        etaE[q] = pe[pq * 32 + q] + be[k2 * 32 + q];
      __syncthreads();
      gj_inv(w0, w1, 32, fcol, &spool[1]);
      matvec(w1, 32, etaE, muE, 32);
      const float qE = block_dot(etaE, muE, 32, red);
      const float zE = 0.5f * (qE + 32.f * LOG2PI - spool[1]);
      const float real = pps[pq] + zE - bz[k2] + bnbs[k2];
      if (tid == 0) {
        int slot = -1;
        if (real > cand[1056]) slot = 0;
        else if (real > cand[1060 + 1056]) slot = 1;
        ctrl[0] = slot;
      }
      __syncthreads();
      const int slot = ctrl[0];
      if (slot >= 0) {
        if (slot == 0) { bcopy(cand + 1060, cand, 1060); __syncthreads(); }
        float* d = cand + slot * 1060;
        bcopy(d, w1, 1024);
        bcopy(d + 1024, muE, 32);
        if (tid == 0) d[1056] = real;
        __syncthreads();
      }
    }
  }

  // emission scores vs NLABELS x OC, then logsumexp over (cand, oc)
  for (int q2 = 0; q2 < G_K; ++q2) {
    for (int l = 0; l < G_NLABELS; ++l) {
      for (int oc = 0; oc < 2; ++oc) {
        const float* om  = omu + ((size_t)l * 2 + oc) * G_DIM;
        const float* och = ocho + ((size_t)l * 2 + oc) * G_DIM;
        for (int idx = tid; idx < 1024; idx += BLK) {
          const int r = idx >> 5, c2 = idx & 31;
          const float dg = (r == c2) ? och[r] * och[r] : 0.f;
          w0[idx] = cand[q2 * 1060 + idx] + dg;
        }
        for (int q = tid; q < 32; q += BLK)
          diff[q] = cand[q2 * 1060 + 1024 + q] - om[q];
        __syncthreads();
        gj_inv(w0, w1, 32, fcol, &spool[2]);
        matvec(w1, 32, diff, sol, 32);
        const float qd = block_dot(diff, sol, 32, red);
        const float sc =
            -0.5f * (32.f * LOG2PI + spool[2] + qd) + cand[q2 * 1060 + 1056];
        if (tid == 0) sc80[(q2 * 2 + oc) * G_NLABELS + l] = sc;
        __syncthreads();
      }
    }
  }
  if (tid < G_NLABELS) {
    float m = -3.0e38f;
    for (int z = 0; z < 4; ++z) m = fmaxf(m, sc80[z * G_NLABELS + tid]);
    float s2 = 0.f;
    for (int z = 0; z < 4; ++z) s2 += expf(sc80[z * G_NLABELS + tid] - m);
    out[((size_t)b * G_MAXLEN + t) * G_NLABELS + tid] = m + logf(s2);
  }
}

// ---------------------------------------------------------------------------
extern "C" void kernel_launch(void* const* d_in, const int* in_sizes, int n_in,
                              void* d_out, int out_size, void* d_ws,
                              size_t ws_size, hipStream_t stream) {
  (void)in_sizes; (void)n_in; (void)out_size; (void)ws_size;
  const int*   sent = (const int*)d_in[0];
  const float* imu  = (const float*)d_in[1];
  const float* icho = (const float*)d_in[2];
  const float* tmu  = (const float*)d_in[3];
  const float* tcho = (const float*)d_in[4];
  const float* omu  = (const float*)d_in[5];
  const float* ocho = (const float*)d_in[6];
  float* hist = (float*)d_ws;   // 16*2*32*2*1057 floats ~= 8.7 MB (L2-resident)
  float* out  = (float*)d_out;

  const size_t scanShm  = (size_t)SCAN_LDS_FLOATS * sizeof(float);
  const size_t mergeShm = (size_t)MERGE_LDS_FLOATS * sizeof(float);

  mgsl_scan_kernel<<<dim3(G_BATCH, 2), dim3(BLK), scanShm, stream>>>(
      sent, imu, icho, tmu, tcho, hist);
  mgsl_merge_kernel<<<dim3(G_BATCH, G_MAXLEN), dim3(BLK), mergeShm, stream>>>(
      sent, imu, icho, omu, ocho, hist, out);
}